// EnergyGatedDelta_3204045603635
// MI455X (gfx1250) — compile-verified
//
#include <hip/hip_runtime.h>

#define H   64
#define VSZ 64
#define BB  256
#define LL  4096
#define TH2 0.16f          // ENERGY_THRESH^2
#define LNE 1e-5f

// workspace layout (float offsets)
#define OFF_KN   0         // 64x64 normalized-k table
#define OFF_V    4096      // 64x64 v table
#define OFF_Q    8192      // 64x64 q table
#define OFF_VN2  12288     // 64    gate thresholds (0.16*||v||^2)
#define OFF_WC   12352     // 64x64 Wrp@Wout
#define OFF_BC   16448     // 64    brp@Wout + bout
#define OFF_READ 16512     // 256x64 read vectors

typedef __attribute__((ext_vector_type(2))) float v2f;
typedef __attribute__((ext_vector_type(8))) float v8f;

// Packed f32 FMA, c += a * broadcast(lo word of b). VOP3P op_sel broadcast.
__device__ __forceinline__ v2f pk_fma_blo(v2f a, v2f b, v2f c) {
    asm("v_pk_fma_f32 %0, %1, %2, %0 op_sel:[0,0,0] op_sel_hi:[1,0,1]"
        : "+v"(c) : "v"(a), "v"(b));
    return c;
}
// Packed f32 FMA, c += a * broadcast(hi word of b).
__device__ __forceinline__ v2f pk_fma_bhi(v2f a, v2f b, v2f c) {
    asm("v_pk_fma_f32 %0, %1, %2, %0 op_sel:[0,1,0] op_sel_hi:[1,1,1]"
        : "+v"(c) : "v"(a), "v"(b));
    return c;
}

struct kpair4 { v2f lo; v2f hi; };   // 16B: 4 consecutive kn values as 2 reg pairs

// ---- wave32 all-lanes sum: DPP8 (xor1/2/4) + DPP16 row_ror:8 (xor8) on the
// VALU path, single ds_swizzle SWAPX16 for the cross-row hop. Much lower
// latency than 5x ds_swizzle on the serial gate chain. ----
#define DPP8SEL(a,b,c,d,e,f,g,h) \
    ((a)|((b)<<3)|((c)<<6)|((d)<<9)|((e)<<12)|((f)<<15)|((g)<<18)|((h)<<21))

template <int SEL>
__device__ __forceinline__ float dpp8_xadd(float x) {
    unsigned p = __builtin_amdgcn_mov_dpp8(__builtin_bit_cast(unsigned, x), SEL);
    return x + __builtin_bit_cast(float, p);
}

__device__ __forceinline__ float wave32_allsum(float x) {
    x = dpp8_xadd<DPP8SEL(1,0,3,2,5,4,7,6)>(x);   // xor 1
    x = dpp8_xadd<DPP8SEL(2,3,0,1,6,7,4,5)>(x);   // xor 2
    x = dpp8_xadd<DPP8SEL(4,5,6,7,0,1,2,3)>(x);   // xor 4
    {   // xor 8: DPP16 row_ror:8 (lane -> lane^8 within each 16-row)
        int s = __builtin_bit_cast(int, x);
        int p = __builtin_amdgcn_update_dpp(s, s, 0x128, 0xf, 0xf, true);
        x = x + __builtin_bit_cast(float, p);
    }
    {   // xor 16: ds_swizzle SWAPX16 (and=0x1f, or=0, xor=0x10)
        int s = __builtin_bit_cast(int, x);
        int p = __builtin_amdgcn_ds_swizzle(s, 0x401f);
        x = x + __builtin_bit_cast(float, p);
    }
    return x;
}

// ---------------------------------------------------------------------------
// Kernel A: collapse encoder + projections into 64-entry tables. 1 block x 64.
// Thread v owns vocab id v end-to-end (LN is per-row, so fully thread-local).
// ---------------------------------------------------------------------------
__global__ __launch_bounds__(64) void egd_tables(
    const float* __restrict__ embed, const float* __restrict__ W1,
    const float* __restrict__ b1,    const float* __restrict__ W2,
    const float* __restrict__ b2,    const float* __restrict__ ln_g,
    const float* __restrict__ ln_b,  const float* __restrict__ Wk,
    const float* __restrict__ Wv,    const float* __restrict__ Wq,
    const float* __restrict__ Wrp,   const float* __restrict__ brp,
    const float* __restrict__ Wout,  const float* __restrict__ bout,
    float* __restrict__ ws)
{
    __shared__ float sh1[64 * 129];          // padded stride: bank (v+j)%64
    const int v = threadIdx.x;

    float e[H];
    for (int i = 0; i < H; ++i) e[i] = embed[v * H + i];

    // h1 = relu(e @ W1 + b1)  (128 wide, staged in LDS to cap VGPR pressure)
    for (int j = 0; j < 2 * H; ++j) {
        float a = b1[j];
        for (int i = 0; i < H; ++i) a = fmaf(e[i], W1[i * (2 * H) + j], a);
        sh1[v * 129 + j] = a > 0.0f ? a : 0.0f;
    }
    // x = e + h1 @ W2 + b2
    float x[H];
    for (int i = 0; i < H; ++i) {
        float a = b2[i];
        for (int j = 0; j < 2 * H; ++j) a = fmaf(sh1[v * 129 + j], W2[j * H + i], a);
        x[i] = e[i] + a;
    }
    // LayerNorm
    float mu = 0.0f;
    for (int i = 0; i < H; ++i) mu += x[i];
    mu *= (1.0f / H);
    float var = 0.0f;
    for (int i = 0; i < H; ++i) { float d = x[i] - mu; var = fmaf(d, d, var); }
    var *= (1.0f / H);
    float rinv = rsqrtf(var + LNE);
    for (int i = 0; i < H; ++i) x[i] = (x[i] - mu) * rinv * ln_g[i] + ln_b[i];

    // k table (normalized) + ||k|| guard
    float kv[H]; float kn2 = 0.0f;
    for (int i = 0; i < H; ++i) {
        float a = 0.0f;
        for (int j = 0; j < H; ++j) a = fmaf(x[j], Wk[j * H + i], a);
        kv[i] = a; kn2 = fmaf(a, a, kn2);
    }
    float kinv = 1.0f / fmaxf(sqrtf(kn2), 1e-12f);
    for (int i = 0; i < H; ++i) ws[OFF_KN + v * H + i] = kv[i] * kinv;

    // v table + squared gate threshold
    float vn2 = 0.0f;
    for (int i = 0; i < H; ++i) {
        float a = 0.0f;
        for (int j = 0; j < H; ++j) a = fmaf(x[j], Wv[j * H + i], a);
        ws[OFF_V + v * H + i] = a; vn2 = fmaf(a, a, vn2);
    }
    ws[OFF_VN2 + v] = TH2 * vn2;

    // q table
    for (int i = 0; i < H; ++i) {
        float a = 0.0f;
        for (int j = 0; j < H; ++j) a = fmaf(x[j], Wq[j * H + i], a);
        ws[OFF_Q + v * H + i] = a;
    }
    // fold readout: Wcomb = Wrp@Wout (row v), bcomb = brp@Wout + bout
    for (int j = 0; j < VSZ; ++j) {
        float a = 0.0f;
        for (int k = 0; k < H; ++k) a = fmaf(Wrp[v * H + k], Wout[k * VSZ + j], a);
        ws[OFF_WC + v * VSZ + j] = a;
    }
    {
        float a = bout[v];
        for (int k = 0; k < H; ++k) a = fmaf(brp[k], Wout[k * VSZ + v], a);
        ws[OFF_BC + v] = a;
    }
}

// ---------------------------------------------------------------------------
// Kernel B: the sequential gated delta scan. 256 blocks x 1 wave32.
// Lane owns rows 2*tid, 2*tid+1 of M as float2 pairs; matvec and rank-1
// update run as V_PK_FMA_F32 (2 FMAs/instr) with op_sel broadcast of the
// shared kn scalar. Gate reduction uses the DPP path above. Tables in LDS;
// no barriers inside the 4095-step dependency chain.
// ---------------------------------------------------------------------------
#define CHUNK 512

__global__ __launch_bounds__(32) void egd_scan(
    const int* __restrict__ seq, const float* __restrict__ ws,
    float* __restrict__ readbuf)
{
    __shared__ __align__(16) float s_kn[VSZ * H];
    __shared__ __align__(16) float s_v [VSZ * H];
    __shared__ float s_vn2[VSZ];
    __shared__ int   s_seq[CHUNK];

    const int tid = threadIdx.x;
    const int b   = blockIdx.x;

    for (int i = tid; i < VSZ * H; i += 32) { s_kn[i] = ws[OFF_KN + i]; s_v[i] = ws[OFF_V + i]; }
    for (int i = tid; i < VSZ;     i += 32) s_vn2[i] = ws[OFF_VN2 + i];
    __syncthreads();

    v2f M[H];                        // M[j] = (row 2*tid [j], row 2*tid+1 [j])
    #pragma unroll
    for (int j = 0; j < H; ++j) M[j] = (v2f){0.0f, 0.0f};

    const int* sb = seq + (size_t)b * LL;

    for (int base = 0; base < LL - 1; base += CHUNK) {
        int cnt = (LL - 1) - base; if (cnt > CHUNK) cnt = CHUNK;
        __syncthreads();
        for (int i = tid; i < cnt; i += 32) s_seq[i] = sb[base + i];
        __syncthreads();

        for (int t = 0; t < cnt; ++t) {
            const int tok = s_seq[t];
            const float* kp = &s_kn[tok * H];

            // issue gate-side loads early so LDS latency hides under the matvec
            const float  thr = s_vn2[tok];
            const float2 vv  = *(const float2*)(&s_v[tok * H + 2 * tid]);

            kpair4 kk[16];
            v2f acc0 = {0.f,0.f}, acc1 = {0.f,0.f}, acc2 = {0.f,0.f}, acc3 = {0.f,0.f};
            #pragma unroll
            for (int c = 0; c < 16; ++c) {
                kk[c] = *(const kpair4*)(kp + 4 * c);        // ds_load_b128
                acc0 = pk_fma_blo(M[4*c+0], kk[c].lo, acc0); // * kn[4c+0]
                acc1 = pk_fma_bhi(M[4*c+1], kk[c].lo, acc1); // * kn[4c+1]
                acc2 = pk_fma_blo(M[4*c+2], kk[c].hi, acc2); // * kn[4c+2]
                acc3 = pk_fma_bhi(M[4*c+3], kk[c].hi, acc3); // * kn[4c+3]
            }
            v2f vp = (acc0 + acc1) + (acc2 + acc3);

            float d0 = vv.x - vp.x, d1 = vv.y - vp.y;
            float err = wave32_allsum(fmaf(d0, d0, d1 * d1));

            float g = (err > thr) ? 1.0f : 0.0f;
            v2f d2 = {d0 * g, d1 * g};
            #pragma unroll
            for (int c = 0; c < 16; ++c) {
                M[4*c+0] = pk_fma_blo(d2, kk[c].lo, M[4*c+0]);
                M[4*c+1] = pk_fma_bhi(d2, kk[c].lo, M[4*c+1]);
                M[4*c+2] = pk_fma_blo(d2, kk[c].hi, M[4*c+2]);
                M[4*c+3] = pk_fma_bhi(d2, kk[c].hi, M[4*c+3]);
            }
        }
    }

    // read = M @ q(last token)
    const int tokL = sb[LL - 1];
    const float* qp = ws + OFF_Q + tokL * H;
    float r0 = 0.f, r1 = 0.f;
    #pragma unroll
    for (int j = 0; j < H; ++j) {
        float qj = qp[j];
        r0 = fmaf(M[j].x, qj, r0);
        r1 = fmaf(M[j].y, qj, r1);
    }
    readbuf[b * H + 2 * tid]     = r0;
    readbuf[b * H + 2 * tid + 1] = r1;
}

// ---------------------------------------------------------------------------
// Kernel C: out = read @ Wcomb + bcomb  via V_WMMA_F32_16X16X4_F32.
// 16 blocks x 1 wave32; each block makes a 16x64 output stripe (4 C tiles),
// K=64 as 16 chained K=4 WMMAs. Fragment layouts per ISA 7.12.2 (f32).
// ---------------------------------------------------------------------------
__global__ __launch_bounds__(32) void egd_out(
    const float* __restrict__ ws, const float* __restrict__ readbuf,
    float* __restrict__ out)
{
    const int lane = threadIdx.x;
    const int rt   = blockIdx.x;         // row tile 0..15
    const int m    = lane & 15;
    const int kh   = (lane >> 4) << 1;   // lanes 0-15 -> K+0/K+1, lanes 16-31 -> K+2/K+3

    const float* Wc = ws + OFF_WC;
    const float* bc = ws + OFF_BC;

    v8f acc[4];
    #pragma unroll
    for (int ct = 0; ct < 4; ++ct) {
        float bias = bc[ct * 16 + m];
        #pragma unroll
        for (int r = 0; r < 8; ++r) acc[ct][r] = bias;
    }

    for (int kk = 0; kk < 16; ++kk) {
        const int k0 = kk * 4;
        v2f a;
        const float* ap = readbuf + (rt * 16 + m) * H + k0 + kh;
        a.x = ap[0]; a.y = ap[1];
        #pragma unroll
        for (int ct = 0; ct < 4; ++ct) {
            v2f bm;
            bm.x = Wc[(k0 + kh)     * VSZ + ct * 16 + m];
            bm.y = Wc[(k0 + kh + 1) * VSZ + ct * 16 + m];
            acc[ct] = __builtin_amdgcn_wmma_f32_16x16x4_f32(
                false, a, false, bm, (short)0, acc[ct], false, false);
        }
    }

    const int rbase = rt * 16 + ((lane >> 4) << 3);   // lanes<16: rows 0-7, else 8-15
    #pragma unroll
    for (int ct = 0; ct < 4; ++ct)
        #pragma unroll
        for (int r = 0; r < 8; ++r)
            out[(rbase + r) * VSZ + ct * 16 + m] = acc[ct][r];
}

// ---------------------------------------------------------------------------
extern "C" void kernel_launch(void* const* d_in, const int* in_sizes, int n_in,
                              void* d_out, int out_size, void* d_ws, size_t ws_size,
                              hipStream_t stream)
{
    const int*   seq   = (const int*)  d_in[0];
    const float* embed = (const float*)d_in[1];
    const float* W1    = (const float*)d_in[2];
    const float* b1    = (const float*)d_in[3];
    const float* W2    = (const float*)d_in[4];
    const float* b2    = (const float*)d_in[5];
    const float* ln_g  = (const float*)d_in[6];
    const float* ln_b  = (const float*)d_in[7];
    const float* Wk    = (const float*)d_in[8];
    const float* Wv    = (const float*)d_in[9];
    const float* Wq    = (const float*)d_in[10];
    const float* Wrp   = (const float*)d_in[11];
    const float* brp   = (const float*)d_in[12];
    const float* Wout  = (const float*)d_in[13];
    const float* bout  = (const float*)d_in[14];

    float* ws  = (float*)d_ws;
    float* out = (float*)d_out;

    egd_tables<<<1, 64, 0, stream>>>(embed, W1, b1, W2, b2, ln_g, ln_b,
                                     Wk, Wv, Wq, Wrp, brp, Wout, bout, ws);
    egd_scan<<<BB, 32, 0, stream>>>(seq, ws, ws + OFF_READ);
    egd_out<<<16, 32, 0, stream>>>(ws, ws + OFF_READ, out);
}